// ThresholdModel_49563922596306
// MI455X (gfx1250) — compile-verified
//
#include <hip/hip_runtime.h>
#include <cstdint>

// Problem geometry (fixed by the reference).
static constexpr int NIMG = 32;
static constexpr int CH   = 3;     // input channels; we read channel 0 only
static constexpr int IMH  = 512;
static constexpr int IMW  = 512;

// Tiling: each block computes a TH x TW output tile.
static constexpr int TH = 32;
static constexpr int TW = 64;
static constexpr int IHAL = TH + 4;   // 36 input rows incl. halo 2
static constexpr int IWAL = 72;       // LDS row stride (288 B, 16B-divisible)
// Row layout in s_img: [0..1 unused][2..3 left halo][4..67 interior (16B-aligned)][68..69 right halo]
static constexpr int MHAL = TH + 2;   // 34 mask rows incl. halo 1
static constexpr int MWAL = TW + 2;   // 66 mask cols incl. halo 1
static constexpr int NTHREADS = 256;  // 8 waves (wave32)

__device__ __forceinline__ int reflect_idx(int i, int n) {
    // jnp.pad 'reflect': -1 -> 1, n -> n-2 (general: -i, 2n-2-i)
    i = (i < 0) ? -i : i;
    i = (i >= n) ? (2 * n - 2 - i) : i;
    return i;
}

__global__ __launch_bounds__(NTHREADS)
void sobel_thresh_dilate_kernel(const float* __restrict__ x,
                                float* __restrict__ out)
{
    __shared__ float s_img[IHAL * IWAL]; // input tile, halo 2, reflect applied
    __shared__ float s_msk[MHAL * MWAL]; // thresholded mask, halo 1

    const int tid     = threadIdx.x;
    const int tile_c0 = blockIdx.x * TW;
    const int tile_r0 = blockIdx.y * TH;
    const int n       = blockIdx.z;

    const float* __restrict__ img = x + (size_t)n * CH * IMH * IMW; // channel 0
    float* __restrict__ o         = out + (size_t)n * IMH * IMW;

    // ---- Phase 1a: async b128 copy of the aligned 64-float interior of each row ----
    // 36 rows x 16 float4 segments = 576 b128 loads per block.
    for (int l = tid; l < IHAL * 16; l += NTHREADS) {
        const int row = l >> 4;          // 0..35
        const int seg = l & 15;          // 0..15
        const int gr  = reflect_idx(tile_r0 + row - 2, IMH); // row reflect only
        const uint32_t ldsoff = (uint32_t)(uintptr_t)(&s_img[row * IWAL + 4 + seg * 4]);
        const uint32_t goff   = (uint32_t)((gr * IMW + tile_c0 + seg * 4) * 4);
        asm volatile("global_load_async_to_lds_b128 %0, %1, %2"
                     :: "v"(ldsoff), "v"(goff), "s"(img)
                     : "memory");
    }
    // ---- Phase 1b: async b32 copy of the 2+2 halo columns (reflected) ----
    for (int l = tid; l < IHAL * 4; l += NTHREADS) {
        const int row = l >> 2;          // 0..35
        const int k   = l & 3;           // 0..3: 0,1 = left halo; 2,3 = right halo
        const int gr  = reflect_idx(tile_r0 + row - 2, IMH);
        const int gcr = (k < 2) ? (tile_c0 - 2 + k) : (tile_c0 + TW + (k - 2));
        const int gc  = reflect_idx(gcr, IMW);
        const int lc  = (k < 2) ? (2 + k) : (TW + 4 + (k - 2));
        const uint32_t ldsoff = (uint32_t)(uintptr_t)(&s_img[row * IWAL + lc]);
        const uint32_t goff   = (uint32_t)((gr * IMW + gc) * 4);
        asm volatile("global_load_async_to_lds_b32 %0, %1, %2"
                     :: "v"(ldsoff), "v"(goff), "s"(img)
                     : "memory");
    }
    asm volatile("s_wait_asynccnt 0" ::: "memory");
    __syncthreads();

    // ---- Phase 2: Sobel magnitude + threshold -> mask tile (halo 1) ----
    // mask (mr,mc) <-> global (tile_r0+mr-1, tile_c0+mc-1);
    // needs img cols (mc-2..mc)+4 = mc+2..mc+4, rows mr..mr+2 in s_img coords.
    for (int l = tid; l < MHAL * MWAL; l += NTHREADS) {
        const int mr = l / MWAL;
        const int mc = l - mr * MWAL;
        const float* p0 = &s_img[(mr + 0) * IWAL + mc + 2];
        const float* p1 = &s_img[(mr + 1) * IWAL + mc + 2];
        const float* p2 = &s_img[(mr + 2) * IWAL + mc + 2];
        const float a = p0[0], b = p0[1], c = p0[2];
        const float d = p1[0],            f = p1[2];
        const float g = p2[0], h = p2[1], i = p2[2];
        // gh: (row above smoothed - row below smoothed)/4 ; gv: (left col - right col)/4
        const float gh = (a + 2.0f * b + c - g - 2.0f * h - i) * 0.25f;
        const float gv = (a + 2.0f * d + g - c - 2.0f * f - i) * 0.25f;
        // mag > 0.5  <=>  gh^2 + gv^2 > 0.25 (both sides nonnegative, sqrt monotone)
        const int gr = tile_r0 + mr - 1;
        const int gc = tile_c0 + mc - 1;
        const bool inimg = (gr >= 0) & (gr < IMH) & (gc >= 0) & (gc < IMW);
        s_msk[l] = (inimg && (gh * gh + gv * gv > 0.25f)) ? 1.0f : 0.0f;
    }
    __syncthreads();

    // ---- Phase 3: 3x3 max dilation (clipped window == outside-as-zero for {0,1}) ----
    const int tx = tid & (TW - 1); // output column within tile: 0..63
    const int ty = tid >> 6;       // 0..3, each handles TH/4 = 8 consecutive rows
    const int r0 = ty * (TH / 4);

    float rm_prev = fmaxf(fmaxf(s_msk[(r0 + 0) * MWAL + tx],
                                s_msk[(r0 + 0) * MWAL + tx + 1]),
                                s_msk[(r0 + 0) * MWAL + tx + 2]);
    float rm_cur  = fmaxf(fmaxf(s_msk[(r0 + 1) * MWAL + tx],
                                s_msk[(r0 + 1) * MWAL + tx + 1]),
                                s_msk[(r0 + 1) * MWAL + tx + 2]);
    #pragma unroll
    for (int k = 0; k < TH / 4; ++k) {
        const int mr = r0 + k + 2;
        const float rm_next = fmaxf(fmaxf(s_msk[mr * MWAL + tx],
                                          s_msk[mr * MWAL + tx + 1]),
                                          s_msk[mr * MWAL + tx + 2]);
        const float v = fmaxf(fmaxf(rm_prev, rm_cur), rm_next);
        o[(size_t)(tile_r0 + r0 + k) * IMW + (tile_c0 + tx)] = v;
        rm_prev = rm_cur;
        rm_cur  = rm_next;
    }
}

extern "C" void kernel_launch(void* const* d_in, const int* in_sizes, int n_in,
                              void* d_out, int out_size, void* d_ws, size_t ws_size,
                              hipStream_t stream) {
    (void)in_sizes; (void)n_in; (void)out_size; (void)d_ws; (void)ws_size;
    const float* x = (const float*)d_in[0];
    float* out = (float*)d_out;
    dim3 grid(IMW / TW, IMH / TH, NIMG);
    dim3 block(NTHREADS);
    sobel_thresh_dilate_kernel<<<grid, block, 0, stream>>>(x, out);
}